// GCN_7576322310639
// MI455X (gfx1250) — compile-verified
//
#include <hip/hip_runtime.h>
#include <hip/hip_bf16.h>
#include <math.h>

#define F_IN 512
#define HDIM 16
#define NCLS 7

typedef float v2f __attribute__((ext_vector_type(2)));
typedef float v8f __attribute__((ext_vector_type(8)));

// ---------------------------------------------------------------------------
// 1) zero a float buffer
__global__ void k_zero(float* __restrict__ p, int n) {
    int i = blockIdx.x * blockDim.x + threadIdx.x;
    if (i < n) p[i] = 0.0f;
}

// 2) in-degree via atomics (deg counted over dst, matching reference)
__global__ void k_deg(const int* __restrict__ dst, float* __restrict__ deg, int e) {
    int i = blockIdx.x * blockDim.x + threadIdx.x;
    if (i < e) atomicAdd(&deg[dst[i]], 1.0f);
}

// 3) dinv = rsqrt(deg + 1)   (self-loop adds 1, so always > 0)
__global__ void k_dinv(float* __restrict__ d, int n) {
    int i = blockIdx.x * blockDim.x + threadIdx.x;
    if (i < n) d[i] = rsqrtf(d[i] + 1.0f);
}

// ---------------------------------------------------------------------------
// 4) GEMM1 via native f32 WMMA: h1 = X @ W1 ;  g1 = acc1 = dinv * h1
//    One wave = one 16-row tile. W1 (512x16 f32 = 32KB) staged in LDS.
//    X is a 205MB single-use stream -> non-temporal loads so it does not
//    evict the L2-resident accumulators / edge index used by later kernels.
__global__ void __launch_bounds__(256) k_gemm1_wmma(
    const float* __restrict__ X, const float* __restrict__ W,
    const float* __restrict__ dinv,
    float* __restrict__ g1, float* __restrict__ acc1, int n)
{
    __shared__ float sW[F_IN * HDIM];   // 32 KB of the WGP's 320 KB LDS
    for (int i = threadIdx.x; i < F_IN * HDIM; i += 256) sW[i] = W[i];
    __syncthreads();

    const int wave = threadIdx.x >> 5;               // 0..7
    const int lane = threadIdx.x & 31;
    const int tile = blockIdx.x * 8 + wave;          // 16-row tile
    const int rowBase = tile * 16;
    if (rowBase >= n) return;                        // wave-uniform: EXEC stays all-1s

    const int hi = lane >> 4;                        // 0 or 1 (K half for A/B)
    const int mn = lane & 15;                        // row (A) / col (B,C)
    const float* __restrict__ xrow = X + (size_t)(rowBase + mn) * F_IN;

    v8f c = {0.f, 0.f, 0.f, 0.f, 0.f, 0.f, 0.f, 0.f};
    #pragma unroll 4
    for (int kb = 0; kb < F_IN / 4; ++kb) {
        const int k = kb * 4 + hi * 2;
        // streaming (non-temporal) b64 load of two consecutive K values
        v2f a = __builtin_nontemporal_load((const v2f*)(xrow + k));
        v2f b;
        b.x = sW[k * HDIM + mn];                     // ds_load_2addr_b32 pairs
        b.y = sW[(k + 1) * HDIM + mn];
        // D = A(16x4) * B(4x16) + C  -> v_wmma_f32_16x16x4_f32
        c = __builtin_amdgcn_wmma_f32_16x16x4_f32(
                false, a, false, b, (short)0, c, false, false);
    }

    // C layout: VGPR r -> M = r (lanes 0-15), M = r+8 (lanes 16-31); N = lane%16
    #pragma unroll
    for (int r = 0; r < 8; ++r) {
        const int row = rowBase + r + hi * 8;
        const float v = c[r] * dinv[row];
        g1[(size_t)row * HDIM + mn]   = v;   // pristine copy for gathers
        acc1[(size_t)row * HDIM + mn] = v;   // accumulator seeded with self-loop term
    }
}

// ---------------------------------------------------------------------------
// 5) edge scatter layer 1: acc1[dst,f] += g1[src,f]; 16 lanes = 16 features
//    g1/acc1 are 6.4MB each -> L2-resident (192MB L2), atomics never hit HBM.
__global__ void k_agg1(const int* __restrict__ src, const int* __restrict__ dst,
                       const float* __restrict__ g1, float* __restrict__ acc1, int e16)
{
    int t = blockIdx.x * blockDim.x + threadIdx.x;
    if (t >= e16) return;
    int ed = t >> 4, f = t & 15;
    int s = src[ed], d = dst[ed];
    atomicAdd(&acc1[(size_t)d * HDIM + f], g1[(size_t)s * HDIM + f]);
}

// 6) finalize layer 1: h1b = relu(dinv * acc1 + b1)
__global__ void k_fin1(const float* __restrict__ acc1, const float* __restrict__ dinv,
                       const float* __restrict__ b1, float* __restrict__ h1b, int n16)
{
    int t = blockIdx.x * blockDim.x + threadIdx.x;
    if (t >= n16) return;
    int row = t >> 4, col = t & 15;
    float v = acc1[t] * dinv[row] + b1[col];
    h1b[t] = v > 0.0f ? v : 0.0f;
}

// ---------------------------------------------------------------------------
// 7) layer 2 GEMM (16x7, per node) + scale:  g2 = acc2 = dinv * (h1b @ W2)
//    padded to 8 columns for aligned addressing
__global__ void k_gemm2(const float* __restrict__ h1b, const float* __restrict__ W2,
                        const float* __restrict__ dinv,
                        float* __restrict__ g2, float* __restrict__ acc2, int n)
{
    int i = blockIdx.x * blockDim.x + threadIdx.x;
    if (i >= n) return;
    float h[HDIM];
    #pragma unroll
    for (int k = 0; k < HDIM; ++k) h[k] = h1b[(size_t)i * HDIM + k];
    const float di = dinv[i];
    #pragma unroll
    for (int c = 0; c < NCLS; ++c) {
        float s = 0.0f;
        #pragma unroll
        for (int k = 0; k < HDIM; ++k) s += h[k] * W2[k * NCLS + c];
        const float v = s * di;
        g2[(size_t)i * 8 + c]   = v;
        acc2[(size_t)i * 8 + c] = v;
    }
    g2[(size_t)i * 8 + 7]   = 0.0f;
    acc2[(size_t)i * 8 + 7] = 0.0f;
}

// 8) edge scatter layer 2: 8 lanes per edge (lane 7 idle pad)
__global__ void k_agg2(const int* __restrict__ src, const int* __restrict__ dst,
                       const float* __restrict__ g2, float* __restrict__ acc2, int e8)
{
    int t = blockIdx.x * blockDim.x + threadIdx.x;
    if (t >= e8) return;
    int ed = t >> 3, f = t & 7;
    if (f >= NCLS) return;
    int s = src[ed], d = dst[ed];
    atomicAdd(&acc2[(size_t)d * 8 + f], g2[(size_t)s * 8 + f]);
}

// 9) finalize layer 2 + log_softmax -> d_out (N x 7, f32); write-once -> NT stores
__global__ void k_fin2(const float* __restrict__ acc2, const float* __restrict__ dinv,
                       const float* __restrict__ b2, float* __restrict__ out, int n)
{
    int i = blockIdx.x * blockDim.x + threadIdx.x;
    if (i >= n) return;
    const float di = dinv[i];
    float o[NCLS];
    float m = -3.402823466e38f;
    #pragma unroll
    for (int c = 0; c < NCLS; ++c) {
        o[c] = acc2[(size_t)i * 8 + c] * di + b2[c];
        m = fmaxf(m, o[c]);
    }
    float s = 0.0f;
    #pragma unroll
    for (int c = 0; c < NCLS; ++c) s += __expf(o[c] - m);
    const float lse = __logf(s);
    #pragma unroll
    for (int c = 0; c < NCLS; ++c)
        __builtin_nontemporal_store(o[c] - m - lse, &out[(size_t)i * NCLS + c]);
}

// ---------------------------------------------------------------------------
static inline int blks(long long work, int bs) { return (int)((work + bs - 1) / bs); }

extern "C" void kernel_launch(void* const* d_in, const int* in_sizes, int n_in,
                              void* d_out, int out_size, void* d_ws, size_t ws_size,
                              hipStream_t stream)
{
    const float* x    = (const float*)d_in[0];
    const int*   ei   = (const int*)d_in[1];
    const float* W1   = (const float*)d_in[2];
    const float* b1   = (const float*)d_in[3];
    const float* W2   = (const float*)d_in[4];
    const float* b2   = (const float*)d_in[5];
    float*       out  = (float*)d_out;

    const int N = in_sizes[0] / F_IN;
    const int E = in_sizes[1] / 2;
    const int* src = ei;
    const int* dst = ei + E;

    // workspace layout (floats): dinv | g1 | acc1 | h1b | g2 | acc2  = 65*N floats
    float* dinv = (float*)d_ws;
    float* g1   = dinv + N;
    float* acc1 = g1   + (size_t)N * HDIM;
    float* h1b  = acc1 + (size_t)N * HDIM;
    float* g2   = h1b  + (size_t)N * HDIM;
    float* acc2 = g2   + (size_t)N * 8;

    const int BS = 256;

    // degree -> dinv
    k_zero<<<blks(N, BS), BS, 0, stream>>>(dinv, N);
    k_deg <<<blks(E, BS), BS, 0, stream>>>(dst, dinv, E);
    k_dinv<<<blks(N, BS), BS, 0, stream>>>(dinv, N);

    // layer 1
    const int tiles = (N + 15) / 16;
    k_gemm1_wmma<<<blks(tiles, 8), BS, 0, stream>>>(x, W1, dinv, g1, acc1, N);
    k_agg1<<<blks((long long)E * HDIM, BS), BS, 0, stream>>>(src, dst, g1, acc1, E * HDIM);
    k_fin1<<<blks((long long)N * HDIM, BS), BS, 0, stream>>>(acc1, dinv, b1, h1b, N * HDIM);

    // layer 2
    k_gemm2<<<blks(N, BS), BS, 0, stream>>>(h1b, W2, dinv, g2, acc2, N);
    k_agg2<<<blks((long long)E * 8, BS), BS, 0, stream>>>(src, dst, g2, acc2, E * 8);
    k_fin2<<<blks(N, BS), BS, 0, stream>>>(acc2, dinv, b2, out, N);
}